// DeltaSynapse_90056874262841
// MI455X (gfx1250) — compile-verified
//
#include <hip/hip_runtime.h>

#define N 2048
#define B 16
#define D 8
#define EC 8            // e-chunks along grid.y
#define WAVES 8         // waves per block
#define TRIPS ((N / EC) / WAVES)   // 32: compile-time trip count -> uniform loop

typedef __attribute__((ext_vector_type(2))) float v2f;
typedef __attribute__((ext_vector_type(8))) float v8f;

// ---------------------------------------------------------------------------
// Prep: A2[e][b][d] = Xd[d][b][e] * (Wshort[d][b][e] + 1), and zero d_out.
// A2 layout [e][b][d] lets each WMMA lane fetch its (k,k+1) A pair with one
// 8-byte-aligned b64 load.
// ---------------------------------------------------------------------------
__global__ __launch_bounds__(256) void dsyn_prep(const float* __restrict__ Wshort,
                                                 const float* __restrict__ Xd,
                                                 float* __restrict__ A2,
                                                 float* __restrict__ out) {
  int idx = blockIdx.x * 256 + threadIdx.x;        // over D*B*N, source-contiguous
  if (idx < D * B * N) {
    int e  = idx & (N - 1);
    int db = idx >> 11;                            // N = 2^11
    int b  = db & (B - 1);
    int d  = db >> 4;                              // B = 2^4
    float a = Xd[idx] * (Wshort[idx] + 1.0f);
    A2[e * (B * D) + b * D + d] = a;
  }
  if (idx < B * N) out[idx] = 0.0f;                // zero accumulator output
}

// ---------------------------------------------------------------------------
// Main: one 16-wide o-tile per blockIdx.x, e-chunk per blockIdx.y.
// Per e: G[16b x 16o] = A_e[16x8] @ Dm_e[8x16] via 2x v_wmma_f32_16x16x4_f32,
// then stream Wlong (non-temporal) and accumulate (P + Q*wl) * G.
// Constant trip count keeps the loop wave-uniform: EXEC is all-ones at every
// WMMA (ISA requirement) and no per-lane exit mask is generated.
// ---------------------------------------------------------------------------
__global__ __launch_bounds__(WAVES * 32) void dsyn_main(const float* __restrict__ W,
                                                        const float* __restrict__ Wlong,
                                                        const float* __restrict__ dm,
                                                        const float* __restrict__ frac,
                                                        const float* __restrict__ sp,
                                                        const float* __restrict__ A2,
                                                        float* __restrict__ out) {
  const int lane = threadIdx.x & 31;
  const int wave = threadIdx.x >> 5;
  const int o0   = blockIdx.x * 16;
  const int n    = lane & 15;          // output column within tile (C/D lane)
  const int hi   = lane >> 4;          // half-wave select
  const int o    = o0 + n;
  const int kb   = hi * 2;             // K base for A/B operand striping
  const size_t NN = (size_t)N * N;

  float iacc[8];
#pragma unroll
  for (int r = 0; r < 8; ++r) iacc[r] = 0.0f;

  const int e_begin = blockIdx.y * (N / EC) + wave;

  for (int t = 0; t < TRIPS; ++t) {    // scalar loop counter, no divergence
    const int e = e_begin + t * WAVES;

    // P/Q per (e, o): b-independent STDP/sign terms
    float spv = sp[e];
    float s   = (spv > 0.0f) ? 1.0f : ((spv < 0.0f) ? -1.0f : 0.0f);
    float w   = W[(size_t)e * N + o];
    float f   = frac[(size_t)e * N + o];
    float sgn = (w > 0.0f) ? s : 0.0f;
    float P   = sgn * w * (1.0f - f);
    float Q   = sgn * f;

    // A operand: lane (m = n) supplies A[m, kb..kb+1] and A[m, kb+4..kb+5]
    const float* ap = A2 + e * (B * D) + n * D;    // L2-resident (A2 = 1 MB)
    v2f a01; a01.x = ap[kb + 0]; a01.y = ap[kb + 1];
    v2f a45; a45.x = ap[kb + 4]; a45.y = ap[kb + 5];

    // B operand: delaymap rows d = kb, kb+1 (vgpr pair 0/1) and kb+4, kb+5
    const size_t dmeo = (size_t)e * N + o;
    v2f b01, b45;
    b01.x = __builtin_nontemporal_load(dm + (size_t)(kb + 0) * NN + dmeo);
    b01.y = __builtin_nontemporal_load(dm + (size_t)(kb + 1) * NN + dmeo);
    b45.x = __builtin_nontemporal_load(dm + (size_t)(kb + 4) * NN + dmeo);
    b45.y = __builtin_nontemporal_load(dm + (size_t)(kb + 5) * NN + dmeo);

    // G = A_e(16x8) @ Dm_e(8x16), chained K=4 + K=4 f32 WMMA
    v8f g = {};
    g = __builtin_amdgcn_wmma_f32_16x16x4_f32(false, a01, false, b01, (short)0, g,
                                              false, false);
    g = __builtin_amdgcn_wmma_f32_16x16x4_f32(false, a45, false, b45, (short)0, g,
                                              false, false);

    // C/D layout: vgpr r, this lane -> (b = r + 8*hi, o). Stream Wlong once (NT).
    const size_t wlbase = (size_t)(hi * 8) * NN + dmeo;
#pragma unroll
    for (int r = 0; r < 8; ++r) {
      float wl = __builtin_nontemporal_load(Wlong + wlbase + (size_t)r * NN);
      iacc[r] = fmaf(P + Q * wl, g[r], iacc[r]);
    }
  }

  // Cross-wave reduction in LDS, then atomic partial-sum into d_out.
  __shared__ float red[WAVES * 256];
#pragma unroll
  for (int r = 0; r < 8; ++r) red[wave * 256 + r * 32 + lane] = iacc[r];
  __syncthreads();

  {
    int j = threadIdx.x;                 // j = r*32 + lane2, covers all 256 slots
    float acc = 0.0f;
#pragma unroll
    for (int wv = 0; wv < WAVES; ++wv) acc += red[wv * 256 + j];
    int r  = j >> 5;
    int l2 = j & 31;
    int bb = r + 8 * (l2 >> 4);
    int oo = o0 + (l2 & 15);
    atomicAdd(&out[bb * N + oo], acc);
  }
}

// ---------------------------------------------------------------------------
// Launch. Input order: W, Wlong, Wshort, Xd, delaymap, STDP_frac, signs_pre.
// ---------------------------------------------------------------------------
extern "C" void kernel_launch(void* const* d_in, const int* in_sizes, int n_in,
                              void* d_out, int out_size, void* d_ws, size_t ws_size,
                              hipStream_t stream) {
  const float* W      = (const float*)d_in[0];
  const float* Wlong  = (const float*)d_in[1];
  const float* Wshort = (const float*)d_in[2];
  const float* Xd     = (const float*)d_in[3];
  const float* dmap   = (const float*)d_in[4];
  const float* frac   = (const float*)d_in[5];
  const float* sp     = (const float*)d_in[6];
  float* out = (float*)d_out;
  float* A2  = (float*)d_ws;                       // D*B*N floats = 1 MB

  dsyn_prep<<<dim3((D * B * N) / 256), dim3(256), 0, stream>>>(Wshort, Xd, A2, out);
  dsyn_main<<<dim3(N / 16, EC), dim3(WAVES * 32), 0, stream>>>(W, Wlong, dmap, frac,
                                                               sp, A2, out);
}